// ChebNetLayer_19095424598406
// MI455X (gfx1250) — compile-verified
//
#include <hip/hip_runtime.h>
#include <hip/hip_bf16.h>
#include <math.h>

// ------------------------------------------------------------------
// ChebNet (K=3) x2 layers for MI455X (gfx1250, wave32).
// Dense GEMMs use V_WMMA_F32_16X16X4_F32 (fp32-exact matrix pipe).
// Sparse propagation uses one wave per edge with coalesced f32 atomics.
// ------------------------------------------------------------------

#define N_NODES 10000
#define DIM     256
#define N_EDGES 320000

typedef float v2f __attribute__((ext_vector_type(2)));
typedef float v8f __attribute__((ext_vector_type(8)));

__device__ __forceinline__ float safef(float v) {
    return isfinite(v) ? v : 0.0f;   // nan_to_num(nan=0, +inf=0, -inf=0)
}

__device__ __forceinline__ void atomic_add_f32(float* p, float v) {
#if defined(__HIP_DEVICE_COMPILE__)
    unsafeAtomicAdd(p, v);           // hardware global_atomic_add_f32
#else
    atomicAdd(p, v);
#endif
}

// ---------------- elementwise helpers ----------------
__global__ __launch_bounds__(256) void k_zero(float* __restrict__ p, int n) {
    int i = blockIdx.x * 256 + threadIdx.x;
    if (i < n) p[i] = 0.0f;
}

__global__ __launch_bounds__(256) void k_safe_copy(float* __restrict__ dst,
                                                   const float* __restrict__ src, int n) {
    int i = blockIdx.x * 256 + threadIdx.x;
    if (i < n) dst[i] = safef(src[i]);
}

__global__ __launch_bounds__(256) void k_neg_copy(float* __restrict__ dst,
                                                  const float* __restrict__ src, int n) {
    int i = blockIdx.x * 256 + threadIdx.x;
    if (i < n) dst[i] = -src[i];
}

// ---------------- edge normalization ----------------
__global__ __launch_bounds__(256) void k_degree(const long long* __restrict__ ei,
                                                const float* __restrict__ w,
                                                float* __restrict__ deg) {
    int e = blockIdx.x * 256 + threadIdx.x;
    if (e >= N_EDGES) return;
    float v = fmaxf(fabsf(safef(w[e])), 1e-6f);
    atomic_add_f32(&deg[(int)ei[e]], v);
}

__global__ __launch_bounds__(256) void k_norm(const long long* __restrict__ ei,
                                              const float* __restrict__ w,
                                              const float* __restrict__ deg,
                                              float* __restrict__ norm) {
    int e = blockIdx.x * 256 + threadIdx.x;
    if (e >= N_EDGES) return;
    float v  = fmaxf(fabsf(safef(w[e])), 1e-6f);
    int   r  = (int)ei[e];
    int   c  = (int)ei[(size_t)N_EDGES + e];
    float dgr = deg[r], dgc = deg[c];
    float dr = dgr > 0.0f ? rsqrtf(dgr) : 0.0f;
    float dc = dgc > 0.0f ? rsqrtf(dgc) : 0.0f;
    norm[e] = -dr * v * dc;
}

// ---------------- sparse propagation: dst[row] += scale*norm[e]*src[col] ----
// One wave (32 lanes) per edge; lanes cover the 256 features in 8 strides.
// Atomic adds per wave hit 32 consecutive f32 addresses -> cacheline-coalesced.
__global__ __launch_bounds__(256) void k_scatter(const float* __restrict__ src,
                                                 const long long* __restrict__ ei,
                                                 const float* __restrict__ norm,
                                                 float* __restrict__ dst,
                                                 float scale) {
    int wid  = (blockIdx.x * 256 + threadIdx.x) >> 5;
    int lane = threadIdx.x & 31;
    if (wid >= N_EDGES) return;
    int   row = (int)ei[wid];
    int   col = (int)ei[(size_t)N_EDGES + wid];
    float nv  = norm[wid] * scale;
    const float* s = src + (size_t)col * DIM;
    float*       d = dst + (size_t)row * DIM;
#pragma unroll
    for (int i = 0; i < 8; ++i) {
        int f = lane + i * 32;
        atomic_add_f32(&d[f], nv * s[f]);
    }
}

// ---------------- fused GEMM: out = bias + A0@W[0] + A1@W[1] + A2@W[2] ------
// One wave computes one 16x16 C tile via V_WMMA_F32_16X16X4_F32 over K=256.
// A fragment layout (32-bit A 16x4): lanes 0-15 -> M=lane, K={0,1};
//                                    lanes 16-31 -> M=lane-16, K={2,3}.
// C layout: VGPR r -> row (r | (lane>=16)*8), col = lane&15.
__global__ __launch_bounds__(256) void k_cheb_gemm(const float* __restrict__ A0,
                                                   const float* __restrict__ A1,
                                                   const float* __restrict__ A2,
                                                   const float* __restrict__ W,   // [3][256][256]
                                                   const float* __restrict__ bias,
                                                   float* __restrict__ out) {
    const int lane = threadIdx.x & 31;
    const int wave = threadIdx.x >> 5;
    const int tile = blockIdx.x * 8 + wave;      // 10000 tiles = 625 M x 16 N
    const int tileN = tile & 15;
    const int tileM = tile >> 4;
    const int nbase = tileN * 16;
    const int mbase = tileM * 16;
    const int ncol  = nbase + (lane & 15);
    const int mrow  = mbase + (lane & 15);
    const int koff  = (lane >> 4) * 2;           // 0 or 2

    v8f acc;
    const float bv = bias[ncol];
#pragma unroll
    for (int r = 0; r < 8; ++r) acc[r] = bv;

    const float* As[3] = {A0, A1, A2};

    for (int k = 0; k < DIM; k += 4) {
#pragma unroll
        for (int c = 0; c < 3; ++c) {
            // A fragment: two consecutive K values -> one 8B vector load
            v2f a = *(const v2f*)(As[c] + (size_t)mrow * DIM + k + koff);
            // B fragment: W[c][k+koff+{0,1}][ncol] (stride-DIM rows)
            const float* Wc = W + (size_t)c * DIM * DIM;
            v2f b;
            b.x = Wc[(size_t)(k + koff)     * DIM + ncol];
            b.y = Wc[(size_t)(k + koff + 1) * DIM + ncol];
            acc = __builtin_amdgcn_wmma_f32_16x16x4_f32(
                false, a, false, b, (short)0, acc, false, false);
        }
    }

#pragma unroll
    for (int r = 0; r < 8; ++r) {
        int row = mbase + r + ((lane >> 4) << 3);
        out[(size_t)row * DIM + ncol] = acc[r];
    }
}

// ---------------- fused LayerNorm + ReLU + residual + safe ------------------
__global__ __launch_bounds__(256) void k_ln_relu_res(const float* __restrict__ gout,
                                                     const float* __restrict__ hres,
                                                     const float* __restrict__ g,
                                                     const float* __restrict__ be,
                                                     float* __restrict__ dst) {
    const int row = blockIdx.x;
    const int tid = threadIdx.x;
    __shared__ float sred[8];
    __shared__ float sbc[2];

    float v = safef(gout[(size_t)row * DIM + tid]);

    // mean
    float s = v;
#pragma unroll
    for (int off = 16; off > 0; off >>= 1) s += __shfl_down(s, off, 32);
    if ((tid & 31) == 0) sred[tid >> 5] = s;
    __syncthreads();
    if (tid == 0) {
        float t = 0.0f;
#pragma unroll
        for (int i = 0; i < 8; ++i) t += sred[i];
        sbc[0] = t * (1.0f / DIM);
    }
    __syncthreads();
    const float mu = sbc[0];
    const float d  = v - mu;

    // variance
    float s2 = d * d;
#pragma unroll
    for (int off = 16; off > 0; off >>= 1) s2 += __shfl_down(s2, off, 32);
    if ((tid & 31) == 0) sred[tid >> 5] = s2;
    __syncthreads();
    if (tid == 0) {
        float t = 0.0f;
#pragma unroll
        for (int i = 0; i < 8; ++i) t += sred[i];
        sbc[1] = t * (1.0f / DIM);
    }
    __syncthreads();
    const float var = sbc[1];

    float y = d * rsqrtf(var + 1e-5f) * g[tid] + be[tid];
    y = safef(y);                       // safe(layer_norm)
    y = fmaxf(y, 0.0f);                 // relu (safe again is identity here)
    y = safef(y + hres[(size_t)row * DIM + tid]);   // safe(residual)
    dst[(size_t)row * DIM + tid] = y;
}

// ---------------------------------------------------------------------------
extern "C" void kernel_launch(void* const* d_in, const int* in_sizes, int n_in,
                              void* d_out, int out_size, void* d_ws, size_t ws_size,
                              hipStream_t stream) {
    (void)in_sizes; (void)n_in; (void)out_size; (void)ws_size;

    const float*     x   = (const float*)d_in[0];
    const long long* ei  = (const long long*)d_in[1];   // int64 edge_index [2,E]
    const float*     ewt = (const float*)d_in[2];
    const float*     Wa[2] = {(const float*)d_in[3], (const float*)d_in[7]};
    const float*     ba[2] = {(const float*)d_in[4], (const float*)d_in[8]};
    const float*     ga[2] = {(const float*)d_in[5], (const float*)d_in[9]};
    const float*     ea[2] = {(const float*)d_in[6], (const float*)d_in[10]};

    const int ND = N_NODES * DIM;   // 2,560,000

    // workspace layout (floats)
    float* ws   = (float*)d_ws;
    float* deg  = ws;                       // N
    float* nrm  = deg + N_NODES;            // E
    float* h    = nrm + N_EDGES;            // N*D
    float* tx1  = h   + ND;                 // N*D
    float* tx2  = tx1 + ND;                 // N*D
    float* gout = tx2 + ND;                 // N*D

    const int eb  = (N_EDGES + 255) / 256;  // 1250 blocks, thread-per-edge
    const int ewb = (N_EDGES * 32 + 255) / 256; // 40000 blocks, wave-per-edge
    const int nb  = (ND + 255) / 256;       // 10000 blocks, thread-per-elem

    // edge normalization
    k_zero<<<(N_NODES + 255) / 256, 256, 0, stream>>>(deg, N_NODES);
    k_degree<<<eb, 256, 0, stream>>>(ei, ewt, deg);
    k_norm<<<eb, 256, 0, stream>>>(ei, ewt, deg, nrm);

    // h = safe(x)
    k_safe_copy<<<nb, 256, 0, stream>>>(h, x, ND);

    for (int layer = 0; layer < 2; ++layer) {
        float* dst = (layer == 1) ? (float*)d_out : h;

        // Tx1 = prop(h)
        k_zero<<<nb, 256, 0, stream>>>(tx1, ND);
        k_scatter<<<ewb, 256, 0, stream>>>(h, ei, nrm, tx1, 1.0f);

        // Tx2 = 2*prop(Tx1) - h   (init with -h, scatter with scale 2)
        k_neg_copy<<<nb, 256, 0, stream>>>(tx2, h, ND);
        k_scatter<<<ewb, 256, 0, stream>>>(tx1, ei, nrm, tx2, 2.0f);

        // gout = bias + h@W0 + Tx1@W1 + Tx2@W2   (fp32 WMMA)
        k_cheb_gemm<<<(N_NODES / 16) * 16 / 8, 256, 0, stream>>>(
            h, tx1, tx2, Wa[layer], ba[layer], gout);

        // dst = safe(relu(safe(LN(gout))) + h)   (in-place into h for layer 0)
        k_ln_relu_res<<<N_NODES, 256, 0, stream>>>(gout, h, ga[layer], ea[layer], dst);
    }
}